// Original_7911329760060
// MI455X (gfx1250) — compile-verified
//
#include <hip/hip_runtime.h>
#include <cstdint>

// Tunables: 256 threads (8 wave32), 16 elements/thread -> 4096-element tiles.
#define TPB        256
#define PER_THREAD 16
#define TILE       (TPB * PER_THREAD)   // 4096
#define NWAVES     (TPB / 32)           // 8
#define LDS_STRIDE 20                   // dwords per thread (16 data + 4 pad): keeps
                                        // 16B chunk alignment, decorrelates banks

__device__ __forceinline__ float clipf(float x, float lo, float hi) {
    return fminf(fmaxf(x, lo), hi);
}

// ---------------------------------------------------------------------------
// Phase 1: per-block product of factor = clip(beta,0,1) + exp(-clip(tau,...))
// ---------------------------------------------------------------------------
__global__ void k_block_prod(const float* __restrict__ beta,
                             const float* __restrict__ tauP,
                             float* __restrict__ part, int n) {
    __shared__ float waveTot[NWAVES];
    const int tid  = threadIdx.x;
    const long long base = (long long)blockIdx.x * TILE;
    const float ef = expf(-clipf(tauP[0], 0.1f, 3.0f));

    float p = 1.0f;
    if (base + TILE <= (long long)n) {
        // lane-contiguous float4 loads: perfectly coalesced
        const float4* src = (const float4*)(beta + base);
        #pragma unroll
        for (int i = 0; i < PER_THREAD / 4; ++i) {
            float4 v = src[tid + i * TPB];
            p *= (clipf(v.x, 0.f, 1.f) + ef);
            p *= (clipf(v.y, 0.f, 1.f) + ef);
            p *= (clipf(v.z, 0.f, 1.f) + ef);
            p *= (clipf(v.w, 0.f, 1.f) + ef);
        }
    } else {
        for (int i = 0; i < PER_THREAD; ++i) {
            long long g = base + tid + (long long)i * TPB;
            if (g < n) p *= (clipf(beta[g], 0.f, 1.f) + ef);
        }
    }
    // wave32 all-lanes product reduction
    #pragma unroll
    for (int d = 16; d >= 1; d >>= 1) p *= __shfl_xor(p, d, 32);
    if ((tid & 31) == 0) waveTot[tid >> 5] = p;
    __syncthreads();
    if (tid == 0) {
        float t = 1.0f;
        #pragma unroll
        for (int w = 0; w < NWAVES; ++w) t *= waveTot[w];
        part[blockIdx.x] = t;
    }
}

// ---------------------------------------------------------------------------
// Phase 2: single-block exclusive cumprod of the block products (in place),
// loops in 4096-wide chunks with a multiplicative carry for generality.
// ---------------------------------------------------------------------------
__global__ void k_scan_partials(float* __restrict__ part, int nb) {
    __shared__ float waveTot[NWAVES];
    __shared__ float wavePre[NWAVES + 1];
    const int tid  = threadIdx.x;
    const int lane = tid & 31;
    const int wid  = tid >> 5;

    float carry = 1.0f;
    for (int base = 0; base < nb; base += TILE) {
        float v, run = 1.0f, incl[PER_THREAD];
        #pragma unroll
        for (int j = 0; j < PER_THREAD; ++j) {
            int c = base + tid * PER_THREAD + j;
            v = (c < nb) ? part[c] : 1.0f;
            run *= v;
            incl[j] = run;
        }
        // wave32 inclusive scan of thread totals
        float wincl = run;
        #pragma unroll
        for (int d = 1; d < 32; d <<= 1) {
            float x = __shfl_up(wincl, d, 32);
            if (lane >= d) wincl *= x;
        }
        float lexcl = __shfl_up(wincl, 1, 32);
        if (lane == 0) lexcl = 1.0f;
        if (lane == 31) waveTot[wid] = wincl;
        __syncthreads();
        if (tid == 0) {
            float r = 1.0f;
            #pragma unroll
            for (int w = 0; w < NWAVES; ++w) { wavePre[w] = r; r *= waveTot[w]; }
            wavePre[NWAVES] = r;   // chunk total
        }
        __syncthreads();
        const float tpre = carry * wavePre[wid] * lexcl;
        #pragma unroll
        for (int j = 0; j < PER_THREAD; ++j) {
            int c = base + tid * PER_THREAD + j;
            if (c < nb) part[c] = tpre * ((j == 0) ? 1.0f : incl[j - 1]);
        }
        carry *= wavePre[NWAVES];
        __syncthreads();   // shared reused next chunk
    }
}

// ---------------------------------------------------------------------------
// Phase 3: async-stage tile to padded LDS, intra-block exclusive cumprod,
// fuse beta * N0 * block_prefix, write out.
// ---------------------------------------------------------------------------
__global__ void k_scan_out(const float* __restrict__ beta,
                           const float* __restrict__ tauP,
                           const float* __restrict__ n0P,
                           const float* __restrict__ blockPre,
                           float* __restrict__ out, int n) {
    __shared__ __align__(16) float tileLds[TPB * LDS_STRIDE];  // 20 KB, padded
    __shared__ float waveTot[NWAVES];
    __shared__ float wavePre[NWAVES];
    const int tid  = threadIdx.x;
    const int lane = tid & 31;
    const int wid  = tid >> 5;
    const long long base = (long long)blockIdx.x * TILE;
    const float ef   = expf(-clipf(tauP[0], 0.1f, 3.0f));
    const float n0   = clipf(n0P[0], 1.0f, 100.0f);
    const float bpre = blockPre[blockIdx.x] * n0;

    // low 32 bits of a generic pointer to LDS == workgroup-relative LDS offset
    const unsigned ldsBase = (unsigned)(uintptr_t)&tileLds[0];
    const bool full = (base + TILE) <= (long long)n;

    if (full) {
        // CDNA5 async global->LDS: each lane copies a 16B chunk (lane-contiguous
        // on the global side) and scatters it to the padded LDS slot so that
        // thread k later reads its 16 *contiguous* scan elements at dword k*20.
        const float* src = beta + base;   // uniform -> SGPR pair (GVS mode)
        #pragma unroll
        for (int i = 0; i < 4; ++i) {
            unsigned c      = (unsigned)tid + (unsigned)i * TPB;      // chunk id
            unsigned ldsOff = ldsBase + ((c >> 2) * LDS_STRIDE + (c & 3u) * 4u) * 4u;
            unsigned goff   = c * 16u;
            asm volatile("global_load_async_to_lds_b128 %0, %1, %2"
                         :: "v"(ldsOff), "v"(goff), "s"(src) : "memory");
        }
        asm volatile("s_wait_asynccnt 0x0" ::: "memory");
    } else {
        for (int e = tid; e < TILE; e += TPB) {
            long long g = base + e;
            tileLds[(e >> 4) * LDS_STRIDE + (e & 15)] = (g < n) ? beta[g] : 0.0f;
        }
    }
    __syncthreads();

    // per-thread serial inclusive cumprod over 16 contiguous elements
    float b[PER_THREAD], incl[PER_THREAD];
    {
        const float4* lp = (const float4*)&tileLds[tid * LDS_STRIDE];
        #pragma unroll
        for (int j = 0; j < 4; ++j) {
            float4 q = lp[j];
            b[4 * j + 0] = clipf(q.x, 0.f, 1.f);
            b[4 * j + 1] = clipf(q.y, 0.f, 1.f);
            b[4 * j + 2] = clipf(q.z, 0.f, 1.f);
            b[4 * j + 3] = clipf(q.w, 0.f, 1.f);
        }
        float run = 1.0f;
        #pragma unroll
        for (int j = 0; j < PER_THREAD; ++j) {
            run *= (b[j] + ef);
            incl[j] = run;
        }
    }

    // wave32 inclusive scan of thread totals, then cross-wave prefix
    float wincl = incl[PER_THREAD - 1];
    #pragma unroll
    for (int d = 1; d < 32; d <<= 1) {
        float x = __shfl_up(wincl, d, 32);
        if (lane >= d) wincl *= x;
    }
    float lexcl = __shfl_up(wincl, 1, 32);
    if (lane == 0) lexcl = 1.0f;
    if (lane == 31) waveTot[wid] = wincl;
    __syncthreads();
    if (tid == 0) {
        float r = 1.0f;
        #pragma unroll
        for (int w = 0; w < NWAVES; ++w) { wavePre[w] = r; r *= waveTot[w]; }
    }
    __syncthreads();

    const float tpre = bpre * wavePre[wid] * lexcl;  // includes N0 & global block prefix

    if (full) {
        float4* dst = (float4*)(out + base + (long long)tid * PER_THREAD);
        #pragma unroll
        for (int j = 0; j < 4; ++j) {
            float4 q;
            int s = 4 * j;
            q.x = b[s + 0] * tpre * ((s + 0 == 0) ? 1.0f : incl[s - 1]);
            q.y = b[s + 1] * tpre * incl[s + 0];
            q.z = b[s + 2] * tpre * incl[s + 1];
            q.w = b[s + 3] * tpre * incl[s + 2];
            dst[j] = q;
        }
    } else {
        #pragma unroll
        for (int j = 0; j < PER_THREAD; ++j) {
            long long g = base + (long long)tid * PER_THREAD + j;
            if (g < n) out[g] = b[j] * tpre * ((j == 0) ? 1.0f : incl[j - 1]);
        }
    }
}

// ---------------------------------------------------------------------------
extern "C" void kernel_launch(void* const* d_in, const int* in_sizes, int n_in,
                              void* d_out, int out_size, void* d_ws, size_t ws_size,
                              hipStream_t stream) {
    (void)n_in; (void)out_size; (void)ws_size;
    const float* beta = (const float*)d_in[0];
    const float* tau  = (const float*)d_in[1];
    const float* N0   = (const float*)d_in[2];
    float* out        = (float*)d_out;
    const int n  = in_sizes[0];
    const int nb = (n + TILE - 1) / TILE;      // 4096 for n = 2^24

    float* part = (float*)d_ws;                // nb floats of scratch

    k_block_prod   <<<nb, TPB, 0, stream>>>(beta, tau, part, n);
    k_scan_partials<<<1,  TPB, 0, stream>>>(part, nb);
    k_scan_out     <<<nb, TPB, 0, stream>>>(beta, tau, N0, part, out, n);
}